// HeterogeneousGNN_48747878809908
// MI455X (gfx1250) — compile-verified
//
#include <hip/hip_runtime.h>

#define CNJ 50000
#define CNM 1000
#define CNS 5000
#define CE1 800000
#define CE2 200000
#define CE3 400000
#define DIM 128
#define NHEAD 4
#define NLAYER 4

typedef float v2f __attribute__((ext_vector_type(2)));
typedef float v8f __attribute__((ext_vector_type(8)));

__device__ __forceinline__ float atomicMaxF(float* addr, float val) {
  if (val >= 0.f) {
    return __int_as_float(atomicMax((int*)addr, __float_as_int(val)));
  } else {
    return __uint_as_float(atomicMin((unsigned int*)addr, __float_as_uint(val)));
  }
}

// ---------------- initial embeddings: Y = relu(X @ W + b), tiny K ----------------
__global__ void k_embed(const float* __restrict__ X, const float* __restrict__ W,
                        const float* __restrict__ b, float* __restrict__ Y,
                        int N, int F) {
  int idx = blockIdx.x * blockDim.x + threadIdx.x;
  if (idx >= N * DIM) return;
  int n = idx / DIM, c = idx % DIM;
  float acc = b[c];
  for (int f = 0; f < F; ++f) acc += X[(size_t)n * F + f] * W[f * DIM + c];
  Y[idx] = fmaxf(acc, 0.f);
}

// ---------------- Y[N,128] = X[N,128] @ W[128,128] + bias, fp32 WMMA ----------------
// block = 256 threads = 8 waves; wave w owns output cols [16w,16w+16); grid.x = ceil(N/16)
__global__ void k_linear128(const float* __restrict__ X, const float* __restrict__ W,
                            const float* __restrict__ bias, float* __restrict__ Y, int N) {
  const int lane = threadIdx.x & 31;
  const int wave = threadIdx.x >> 5;   // 0..7 -> column tile
  const int m16  = lane & 15;
  const int kp   = lane >> 4;          // 0 or 1 : K-pair selector
  const int row  = blockIdx.x * 16 + m16;
  const int rowc = row < N ? row : (N - 1);   // clamp: D-row m depends only on A-row m
  const int col  = wave * 16 + m16;           // B/C/D n-index = lane&15
  const float* __restrict__ xr = X + (size_t)rowc * DIM;

  v8f acc = {0.f, 0.f, 0.f, 0.f, 0.f, 0.f, 0.f, 0.f};
#pragma unroll
  for (int k0 = 0; k0 < DIM; k0 += 4) {
    const int ka = k0 + 2 * kp;
    v2f a, b;
    a.x = xr[ka];
    a.y = xr[ka + 1];
    b.x = W[(size_t)ka * DIM + col];
    b.y = W[(size_t)(ka + 1) * DIM + col];
    acc = __builtin_amdgcn_wmma_f32_16x16x4_f32(false, a, false, b, (short)0, acc,
                                                false, false);
  }
  const float bv = bias[col];
#pragma unroll
  for (int v = 0; v < 8; ++v) {
    const int mo = v + 8 * kp;                 // C/D layout: M = vgpr + 8*(lane>=16)
    const int orow = blockIdx.x * 16 + mo;
    if (orow < N) Y[(size_t)orow * DIM + col] = acc[v] + bv;
  }
}

// ---------------- per-GAT segment-state init ----------------
__global__ void k_gat_init(float* __restrict__ outb, float* __restrict__ lmax,
                           float* __restrict__ lsum, int Ndst) {
  int idx = blockIdx.x * blockDim.x + threadIdx.x;
  if (idx >= Ndst * DIM) return;
  outb[idx] = 0.f;
  int c = idx % DIM;
  if (c < NHEAD) {
    int n = idx / DIM;
    lmax[n * NHEAD + c] = -__builtin_huge_valf();
    lsum[n * NHEAD + c] = 0.f;
  }
}

__global__ void k_zero(float* __restrict__ p, int n) {
  int i = blockIdx.x * blockDim.x + threadIdx.x;
  if (i < n) p[i] = 0.f;
}

// ---------------- edge pass 1: logits + segment max. one wave per edge ----------------
__global__ void k_edge_logits(const int* __restrict__ srcIdx, const int* __restrict__ dstIdx,
                              const float* __restrict__ ea, int edim,
                              const float* __restrict__ We,  // [edim,128]
                              const float* __restrict__ xl, const float* __restrict__ xr,
                              const float* __restrict__ att, // [128] flat (H*C)
                              float* __restrict__ logits, float* __restrict__ lmax, int E) {
  const int e = blockIdx.x * (blockDim.x >> 5) + (threadIdx.x >> 5);
  if (e >= E) return;
  const int lane = threadIdx.x & 31;
  const int s = srcIdx[e], d = dstIdx[e];
  float ef[3];
  for (int f = 0; f < edim; ++f) ef[f] = ea[(size_t)e * edim + f];
  const int c0 = lane * 4;  // 4 channels per lane; head = lane/8
  float partial = 0.f;
#pragma unroll
  for (int j = 0; j < 4; ++j) {
    const int c = c0 + j;
    float ev = 0.f;
    for (int f = 0; f < edim; ++f) ev += ef[f] * We[f * DIM + c];
    float mv = xl[(size_t)s * DIM + c] + xr[(size_t)d * DIM + c] + ev;
    mv = mv > 0.f ? mv : 0.2f * mv;  // leaky_relu(0.2)
    partial += mv * att[c];
  }
  // reduce over the 8 lanes of one head (xor masks 1/2/4 stay inside the group)
  partial += __shfl_xor(partial, 1);
  partial += __shfl_xor(partial, 2);
  partial += __shfl_xor(partial, 4);
  if ((lane & 7) == 0) {
    const int h = lane >> 3;
    logits[(size_t)e * NHEAD + h] = partial;
    atomicMaxF(&lmax[(size_t)d * NHEAD + h], partial);
  }
}

// ---------------- edge pass 2: p = exp(l - max), denom += p, out += xl[src]*p ----------------
__global__ void k_edge_scatter(const int* __restrict__ srcIdx, const int* __restrict__ dstIdx,
                               const float* __restrict__ xl, const float* __restrict__ logits,
                               const float* __restrict__ lmax, float* __restrict__ lsum,
                               float* __restrict__ outb, int E) {
  const int e = blockIdx.x * (blockDim.x >> 5) + (threadIdx.x >> 5);
  if (e >= E) return;
  const int lane = threadIdx.x & 31;
  const int s = srcIdx[e], d = dstIdx[e];
  const int h = lane >> 3;
  const float p = __expf(logits[(size_t)e * NHEAD + h] - lmax[(size_t)d * NHEAD + h]);
  if ((lane & 7) == 0) atomicAdd(&lsum[(size_t)d * NHEAD + h], p);
  const int c0 = lane * 4;
#pragma unroll
  for (int j = 0; j < 4; ++j) {
    const int c = c0 + j;
    atomicAdd(&outb[(size_t)d * DIM + c], xl[(size_t)s * DIM + c] * p);
  }
}

// ---------------- finalize: accum += 0.5*(out/denom + bo) ----------------
__global__ void k_finalize(const float* __restrict__ outb, const float* __restrict__ lsum,
                           const float* __restrict__ bo, float* __restrict__ accum, int Ndst) {
  int idx = blockIdx.x * blockDim.x + threadIdx.x;
  if (idx >= Ndst * DIM) return;
  int n = idx / DIM, c = idx % DIM, h = c >> 5;
  float denom = fmaxf(lsum[n * NHEAD + h], 1e-16f);
  accum[idx] += 0.5f * (outb[idx] / denom + bo[c]);
}

// ---------------- h = LN(h + relu(new)); one wave per node row ----------------
__global__ void k_update(float* __restrict__ hx, const float* __restrict__ newb,
                         const float* __restrict__ g, const float* __restrict__ b, int N) {
  const int n = blockIdx.x * (blockDim.x >> 5) + (threadIdx.x >> 5);
  if (n >= N) return;
  const int lane = threadIdx.x & 31;
  float v[4];
  float s = 0.f;
#pragma unroll
  for (int j = 0; j < 4; ++j) {
    const int c = lane * 4 + j;
    float x = hx[(size_t)n * DIM + c] + fmaxf(newb[(size_t)n * DIM + c], 0.f);
    v[j] = x;
    s += x;
  }
#pragma unroll
  for (int o = 1; o < 32; o <<= 1) s += __shfl_xor(s, o);
  const float mu = s * (1.f / 128.f);
  float vs = 0.f;
#pragma unroll
  for (int j = 0; j < 4; ++j) {
    float d = v[j] - mu;
    vs += d * d;
  }
#pragma unroll
  for (int o = 1; o < 32; o <<= 1) vs += __shfl_xor(vs, o);
  const float inv = rsqrtf(vs * (1.f / 128.f) + 1e-5f);
#pragma unroll
  for (int j = 0; j < 4; ++j) {
    const int c = lane * 4 + j;
    hx[(size_t)n * DIM + c] = (v[j] - mu) * inv * g[c] + b[c];
  }
}

// =====================================================================================
static inline int ceil_div(int a, int b) { return (a + b - 1) / b; }

static void gat_call(const float* hsrc, int Nsrc, const float* hdst, int Ndst,
                     const int* sIdx, const int* dIdx, int E,
                     const float* ea, int edim,
                     const float* Wl, const float* bl, const float* Wr, const float* br,
                     const float* We, const float* att, const float* bo,
                     float* xl, float* xr, float* logits, float* lmax, float* lsum,
                     float* outb, float* accum, hipStream_t stream) {
  k_linear128<<<ceil_div(Nsrc, 16), 256, 0, stream>>>(hsrc, Wl, bl, xl, Nsrc);
  k_linear128<<<ceil_div(Ndst, 16), 256, 0, stream>>>(hdst, Wr, br, xr, Ndst);
  k_gat_init<<<ceil_div(Ndst * DIM, 256), 256, 0, stream>>>(outb, lmax, lsum, Ndst);
  k_edge_logits<<<ceil_div(E, 8), 256, 0, stream>>>(sIdx, dIdx, ea, edim, We, xl, xr, att,
                                                    logits, lmax, E);
  k_edge_scatter<<<ceil_div(E, 8), 256, 0, stream>>>(sIdx, dIdx, xl, logits, lmax, lsum,
                                                     outb, E);
  k_finalize<<<ceil_div(Ndst * DIM, 256), 256, 0, stream>>>(outb, lsum, bo, accum, Ndst);
}

extern "C" void kernel_launch(void* const* d_in, const int* in_sizes, int n_in,
                              void* d_out, int out_size, void* d_ws, size_t ws_size,
                              hipStream_t stream) {
  const float* xj = (const float*)d_in[0];
  const float* xm = (const float*)d_in[1];
  const float* xs = (const float*)d_in[2];
  const int* src_cp = (const int*)d_in[3];
  const int* dst_cp = (const int*)d_in[4];
  const float* ea_cp = (const float*)d_in[5];
  const int* src_hs = (const int*)d_in[6];
  const int* dst_hs = (const int*)d_in[7];
  const float* ea_hs = (const float*)d_in[8];
  const int* src_rs = (const int*)d_in[9];
  const int* dst_rs = (const int*)d_in[10];
  const float* ea_rs = (const float*)d_in[11];
  const float* Wj = (const float*)d_in[12];
  const float* bj = (const float*)d_in[13];
  const float* Wm = (const float*)d_in[14];
  const float* bm = (const float*)d_in[15];
  const float* Ws = (const float*)d_in[16];
  const float* bs = (const float*)d_in[17];
  const float* Wl = (const float*)d_in[18];
  const float* bl = (const float*)d_in[19];
  const float* Wr = (const float*)d_in[20];
  const float* br = (const float*)d_in[21];
  const float* We3 = (const float*)d_in[22];
  const float* We1 = (const float*)d_in[23];
  const float* att = (const float*)d_in[24];
  const float* bo = (const float*)d_in[25];
  const float* ln_g = (const float*)d_in[26];
  const float* ln_b = (const float*)d_in[27];

  // hidden states live directly in d_out (output = concat(hj, hm, hs))
  float* hj = (float*)d_out;
  float* hm = hj + (size_t)CNJ * DIM;
  float* hs = hm + (size_t)CNM * DIM;

  // carve workspace
  char* wp = (char*)d_ws;
  auto carve = [&](size_t nfloats) {
    float* p = (float*)wp;
    wp += ((nfloats * sizeof(float)) + 255) & ~(size_t)255;
    return p;
  };
  float* xl     = carve((size_t)CNJ * DIM);      // max Nsrc
  float* xr     = carve((size_t)CNJ * DIM);      // max Ndst
  float* logits = carve((size_t)CE1 * NHEAD);    // max E
  float* lmax   = carve((size_t)CNJ * NHEAD);
  float* lsum   = carve((size_t)CNJ * NHEAD);
  float* outb   = carve((size_t)CNJ * DIM);
  float* newj   = carve((size_t)CNJ * DIM);
  float* newm   = carve((size_t)CNM * DIM);
  float* news   = carve((size_t)CNS * DIM);

  // initial embeddings
  k_embed<<<ceil_div(CNJ * DIM, 256), 256, 0, stream>>>(xj, Wj, bj, hj, CNJ, 5);
  k_embed<<<ceil_div(CNM * DIM, 256), 256, 0, stream>>>(xm, Wm, bm, hm, CNM, 4);
  k_embed<<<ceil_div(CNS * DIM, 256), 256, 0, stream>>>(xs, Ws, bs, hs, CNS, 2);

  for (int i = 0; i < NLAYER; ++i) {
    auto WlP = [&](int r) { return Wl + (size_t)(r * NLAYER + i) * DIM * DIM; };
    auto WrP = [&](int r) { return Wr + (size_t)(r * NLAYER + i) * DIM * DIM; };
    auto blP = [&](int r) { return bl + (size_t)(r * NLAYER + i) * DIM; };
    auto brP = [&](int r) { return br + (size_t)(r * NLAYER + i) * DIM; };
    auto We3P = [&](int a) { return We3 + (size_t)(a * NLAYER + i) * 3 * DIM; };
    auto We1P = [&](int a) { return We1 + (size_t)(a * NLAYER + i) * 1 * DIM; };
    auto attP = [&](int r) { return att + (size_t)(r * NLAYER + i) * DIM; };
    auto boP  = [&](int r) { return bo + (size_t)(r * NLAYER + i) * DIM; };
    auto lgP  = [&](int t) { return ln_g + (size_t)(t * NLAYER + i) * DIM; };
    auto lbP  = [&](int t) { return ln_b + (size_t)(t * NLAYER + i) * DIM; };

    k_zero<<<ceil_div(CNJ * DIM, 256), 256, 0, stream>>>(newj, CNJ * DIM);
    k_zero<<<ceil_div(CNM * DIM, 256), 256, 0, stream>>>(newm, CNM * DIM);
    k_zero<<<ceil_div(CNS * DIM, 256), 256, 0, stream>>>(news, CNS * DIM);

    // r0: jobs -> machines (cp)
    gat_call(hj, CNJ, hm, CNM, src_cp, dst_cp, CE1, ea_cp, 3,
             WlP(0), blP(0), WrP(0), brP(0), We3P(0), attP(0), boP(0),
             xl, xr, logits, lmax, lsum, outb, newm, stream);
    // r3: setups -> machines (rev_hs)
    gat_call(hs, CNS, hm, CNM, dst_hs, src_hs, CE2, ea_hs, 1,
             WlP(3), blP(3), WrP(3), brP(3), We1P(1), attP(3), boP(3),
             xl, xr, logits, lmax, lsum, outb, newm, stream);
    // r1: machines -> jobs (rev_cp)
    gat_call(hm, CNM, hj, CNJ, dst_cp, src_cp, CE1, ea_cp, 3,
             WlP(1), blP(1), WrP(1), brP(1), We3P(1), attP(1), boP(1),
             xl, xr, logits, lmax, lsum, outb, newj, stream);
    // r5: setups -> jobs (rev_rs)
    gat_call(hs, CNS, hj, CNJ, dst_rs, src_rs, CE3, ea_rs, 1,
             WlP(5), blP(5), WrP(5), brP(5), We1P(3), attP(5), boP(5),
             xl, xr, logits, lmax, lsum, outb, newj, stream);
    // r2: machines -> setups (hs)
    gat_call(hm, CNM, hs, CNS, src_hs, dst_hs, CE2, ea_hs, 1,
             WlP(2), blP(2), WrP(2), brP(2), We1P(0), attP(2), boP(2),
             xl, xr, logits, lmax, lsum, outb, news, stream);
    // r4: jobs -> setups (rs)
    gat_call(hj, CNJ, hs, CNS, src_rs, dst_rs, CE3, ea_rs, 1,
             WlP(4), blP(4), WrP(4), brP(4), We1P(2), attP(4), boP(4),
             xl, xr, logits, lmax, lsum, outb, news, stream);

    // updates: h = LN(h + relu(new))
    k_update<<<ceil_div(CNJ, 8), 256, 0, stream>>>(hj, newj, lgP(0), lbP(0), CNJ);
    k_update<<<ceil_div(CNM, 8), 256, 0, stream>>>(hm, newm, lgP(1), lbP(1), CNM);
    k_update<<<ceil_div(CNS, 8), 256, 0, stream>>>(hs, news, lgP(2), lbP(2), CNS);
  }
  (void)in_sizes; (void)n_in; (void)out_size; (void)ws_size;
}